// OverlappingLoss_44281112821855
// MI455X (gfx1250) — compile-verified
//
#include <hip/hip_runtime.h>

#define C_CLUSTERS 256
#define DIM 32
#define NPAIRS (C_CLUSTERS * C_CLUSTERS)

typedef float v2f __attribute__((ext_vector_type(2)));
typedef float v8f __attribute__((ext_vector_type(8)));

// Wave-uniform broadcast from a compile-time-constant lane: lowers to
// v_readlane_b32 into an SGPR (no LDS traffic, no dscnt waits).
__device__ __forceinline__ float rlane(float v, int l) {
  return __int_as_float(__builtin_amdgcn_readlane(__float_as_int(v), l));
}

// ---------------------------------------------------------------------------
// Kernel 1: one wave per cluster. sigma[c] = exp(S[c]) * exp(S[c])^T using
// V_WMMA_F32_16X16X4_F32 (exact fp32 WMMA). 2x2 output tiles of 16x16, K=32
// split into 8 steps of K=4.
//
// A-operand layout (16x4 MxK, 2 VGPRs/lane): lane L, reg r -> M = L&15,
// K = 2*(L>>4) + r.  B-operand (4x16 KxN) mirrors with N = L&15. Since
// B = A^T tile, the per-lane data of B-frag(n,k) equals A-frag(n,k), so one
// fragment set F[tile][k] serves both operands.
// ---------------------------------------------------------------------------
__global__ __launch_bounds__(256) void sigma_wmma_kernel(
    const float* __restrict__ sigma_inv, float* __restrict__ sigma) {
  const int c = (blockIdx.x * blockDim.x + threadIdx.x) >> 5;  // cluster
  const int lane = threadIdx.x & 31;
  if (c >= C_CLUSTERS) return;  // exact grid: never taken
  const float* __restrict__ Ainv = sigma_inv + (size_t)c * DIM * DIM;
  const int half = lane >> 4;  // 0..1
  const int lrow = lane & 15;  // 0..15

  // Load + exp the 16 fragments (2 row-tiles x 8 K-steps), 2 floats per lane.
  // Inputs are N(0,1): __expf (v_exp_f32) is plenty accurate, no range fixups.
  v2f F[2][8];
#pragma unroll
  for (int t = 0; t < 2; ++t) {
    const float2* row = (const float2*)(Ainv + (16 * t + lrow) * DIM);
#pragma unroll
    for (int k = 0; k < 8; ++k) {
      float2 v = row[2 * k + half];  // cols 4k+2*half .. +1
      v2f f;
      f.x = __expf(v.x);
      f.y = __expf(v.y);
      F[t][k] = f;
    }
  }

  v8f acc00 = {}, acc01 = {}, acc10 = {}, acc11 = {};
#pragma unroll
  for (int k = 0; k < 8; ++k) {
    acc00 = __builtin_amdgcn_wmma_f32_16x16x4_f32(false, F[0][k], false, F[0][k],
                                                  (short)0, acc00, false, false);
    acc01 = __builtin_amdgcn_wmma_f32_16x16x4_f32(false, F[0][k], false, F[1][k],
                                                  (short)0, acc01, false, false);
    acc10 = __builtin_amdgcn_wmma_f32_16x16x4_f32(false, F[1][k], false, F[0][k],
                                                  (short)0, acc10, false, false);
    acc11 = __builtin_amdgcn_wmma_f32_16x16x4_f32(false, F[1][k], false, F[1][k],
                                                  (short)0, acc11, false, false);
  }

  // C/D layout: reg r, lane L -> row = 16m + r + 8*(L>>4), col = 16n + (L&15).
  float* __restrict__ out = sigma + (size_t)c * DIM * DIM;
#pragma unroll
  for (int r = 0; r < 8; ++r) {
    const int row0 = r + 8 * half;
    const int col0 = lrow;
    out[(row0)*DIM + (col0)] = acc00[r];
    out[(row0)*DIM + (col0 + 16)] = acc01[r];
    out[(row0 + 16) * DIM + (col0)] = acc10[r];
    out[(row0 + 16) * DIM + (col0 + 16)] = acc11[r];
  }
}

// ---------------------------------------------------------------------------
// Kernel 2: one wave per (i,j) pair, upper triangle only (i <= j; symmetric
// pairs weighted x2). Lane l holds row l of M = 0.5(Si+Sj) in 32 VGPRs.
// Wave-parallel in-place Cholesky with v_readlane broadcasts, then forward
// solve L y = dmu; quad = ||y||^2; logdet = sum log(pivots). The diagonal
// lane of each L-column stores 1/Lkk so the solve is division-free.
// ---------------------------------------------------------------------------
__global__ __launch_bounds__(256) void pair_chol_kernel(
    const float* __restrict__ mu, const float* __restrict__ sigma,
    float* __restrict__ sPair, float* __restrict__ diagLogdet) {
  const int wave = (blockIdx.x * blockDim.x + threadIdx.x) >> 5;
  const int lane = threadIdx.x & 31;
  const int i = wave >> 8;
  const int j = wave & (C_CLUSTERS - 1);
  if (i > j) return;  // (i,j) uniform per wave: whole-wave exit, no divergence

  const float* __restrict__ Si = sigma + (size_t)i * DIM * DIM + lane * DIM;
  const float* __restrict__ Sj = sigma + (size_t)j * DIM * DIM + lane * DIM;
  __builtin_prefetch(Si, 0, 0);  // global_prefetch_b8
  __builtin_prefetch(Sj, 0, 0);

  float m[DIM];
  {
    const float4* si4 = (const float4*)Si;
    const float4* sj4 = (const float4*)Sj;
#pragma unroll
    for (int q = 0; q < 8; ++q) {
      float4 a = si4[q];
      float4 b4 = sj4[q];
      m[4 * q + 0] = 0.5f * (a.x + b4.x);
      m[4 * q + 1] = 0.5f * (a.y + b4.y);
      m[4 * q + 2] = 0.5f * (a.z + b4.z);
      m[4 * q + 3] = 0.5f * (a.w + b4.w);
    }
  }
  float b = mu[i * DIM + lane] - mu[j * DIM + lane];

  // Cholesky: lane l = row l. After step k, m[k] holds L[l][k] for l > k and
  // 1/L[k][k] at lane k. Pivots are positive (SPD): __logf/rsqrtf are safe.
  float logdet = 0.0f;
#pragma unroll
  for (int k = 0; k < DIM; ++k) {
    const float Mkk = rlane(m[k], k);  // updated pivot (valid at lane k)
    logdet += __logf(Mkk);             // = 2*log(Lkk), uniform on all lanes
    const float inv = rsqrtf(Mkk);     // 1/Lkk
    const float Lik = m[k] * inv;      // valid for lanes >= k
    m[k] = (lane == k) ? inv : Lik;    // stash reciprocal diagonal
#pragma unroll
    for (int jj = k + 1; jj < DIM; ++jj) {
      m[jj] = fmaf(-Lik, rlane(Lik, jj), m[jj]);
    }
  }

  // Forward solve L y = b ; quad = sum y_k^2 (uniform). Division-free:
  // y_k = b_k * (1/Lkk), both via readlane.
  float quad = 0.0f;
#pragma unroll
  for (int k = 0; k < DIM; ++k) {
    const float yk = rlane(b, k) * rlane(m[k], k);
    quad = fmaf(yk, yk, quad);
    if (lane > k) b = fmaf(-m[k], yk, b);
  }

  if (lane == 0) {
    const float w = (i == j) ? 1.0f : 2.0f;  // symmetry weight
    sPair[wave] = w * (0.125f * quad + 0.5f * logdet);
    if (i == j) diagLogdet[i] = logdet;  // sig_pair(i,i) = sigma_i
  }
}

// ---------------------------------------------------------------------------
// Kernel 3: deterministic reduction (fixed strided partials + LDS tree).
// Only upper-triangle entries of sPair were written; skip the rest.
// out = -( S1 - 0.5*C*S2 ) / C^2
// ---------------------------------------------------------------------------
__global__ __launch_bounds__(256) void reduce_kernel(
    const float* __restrict__ sPair, const float* __restrict__ diagLogdet,
    float* __restrict__ out) {
  __shared__ float sh1[256];
  __shared__ float sh2[256];
  const int t = threadIdx.x;
  float s1 = 0.0f;
  for (int p = t; p < NPAIRS; p += 256) {
    const int pi = p >> 8;
    const int pj = p & (C_CLUSTERS - 1);
    if (pi <= pj) s1 += sPair[p];
  }
  sh1[t] = s1;
  sh2[t] = diagLogdet[t];
  __syncthreads();
  for (int off = 128; off > 0; off >>= 1) {
    if (t < off) {
      sh1[t] += sh1[t + off];
      sh2[t] += sh2[t + off];
    }
    __syncthreads();
  }
  if (t == 0) {
    const float total = sh1[0] - 0.5f * (float)C_CLUSTERS * sh2[0];
    out[0] = -total / (float)NPAIRS;
  }
}

// ---------------------------------------------------------------------------
extern "C" void kernel_launch(void* const* d_in, const int* in_sizes, int n_in,
                              void* d_out, int out_size, void* d_ws,
                              size_t ws_size, hipStream_t stream) {
  const float* mu = (const float*)d_in[0];         // [256,32]
  const float* sigma_inv = (const float*)d_in[1];  // [256,32,32]

  float* sigma = (float*)d_ws;                    // 256*32*32 = 262144 floats
  float* sPair = sigma + C_CLUSTERS * DIM * DIM;  // 65536 floats
  float* diag = sPair + NPAIRS;                   // 256 floats

  // 256 waves (one per cluster): 32 blocks x 256 threads
  sigma_wmma_kernel<<<32, 256, 0, stream>>>(sigma_inv, sigma);
  // 65536 waves (one per pair; lower triangle exits immediately)
  pair_chol_kernel<<<NPAIRS / 8, 256, 0, stream>>>(mu, sigma, sPair, diag);
  // single-block deterministic reduction
  reduce_kernel<<<1, 256, 0, stream>>>(sPair, diag, (float*)d_out);
}